// CriticGNN_10385230921848
// MI455X (gfx1250) — compile-verified
//
#include <hip/hip_runtime.h>
#include <hip/hip_bf16.h>
#include <limits.h>

typedef __attribute__((ext_vector_type(2))) float v2f;
typedef __attribute__((ext_vector_type(8))) float v8f;

#define EPSM 1e-7f

__device__ __forceinline__ int f2key(float f) {
  int i = __float_as_int(f);
  return i >= 0 ? i : (i ^ 0x7fffffff);
}
__device__ __forceinline__ float key2f(int i) {
  return __int_as_float(i >= 0 ? i : (i ^ 0x7fffffff));
}

__device__ __forceinline__ v8f wmma4(v2f a, v2f b, v8f c) {
  // D = A(16x4) * B(4x16) + C, fp32
  return __builtin_amdgcn_wmma_f32_16x16x4_f32(
      false, a, false, b, (short)0, c, false, false);
}

// ---------------- node encoder: h = x @ node_w + node_b  (K=64) ----------------
__global__ void k_node_encode(const float* __restrict__ x, const float* __restrict__ w,
                              const float* __restrict__ bias, float* __restrict__ h,
                              int ntiles) {
  int gw = (int)((blockIdx.x * blockDim.x + threadIdx.x) >> 5);
  if (gw >= ntiles) return;                 // wave-uniform
  int lane = threadIdx.x & 31;
  int lrow = lane & 15;
  int half = lane >> 4;
  int col = lane & 15;
  int row0 = gw << 4;
  v8f c = {0.f, 0.f, 0.f, 0.f, 0.f, 0.f, 0.f, 0.f};
#pragma unroll
  for (int kk = 0; kk < 64; kk += 4) {
    int ka = kk + 2 * half;
    v2f a, b;
    const float* xp = x + (size_t)(row0 + lrow) * 64 + ka;
    a.x = xp[0];
    a.y = xp[1];
    b.x = w[(size_t)ka * 16 + col];
    b.y = w[(size_t)(ka + 1) * 16 + col];
    c = wmma4(a, b, c);
  }
  float bv = bias[col];
#pragma unroll
  for (int v = 0; v < 8; ++v) {
    int m = v + 8 * half;
    h[(size_t)(row0 + m) * 16 + col] = c[v] + bv;
  }
}

// ---------------- y = relu(LN(h)) per node (layers 1..3) ----------------
__global__ void k_prep_y(const float* __restrict__ h, const float* __restrict__ g,
                         const float* __restrict__ be, float* __restrict__ y, int n) {
  int i = blockIdx.x * blockDim.x + threadIdx.x;
  if (i >= n) return;
  float r[16];
  const float4* hp = (const float4*)(h + (size_t)i * 16);
#pragma unroll
  for (int q = 0; q < 4; ++q) {
    float4 v = hp[q];
    r[4 * q] = v.x; r[4 * q + 1] = v.y; r[4 * q + 2] = v.z; r[4 * q + 3] = v.w;
  }
  float sm = 0.f, sq = 0.f;
#pragma unroll
  for (int f = 0; f < 16; ++f) { sm += r[f]; sq += r[f] * r[f]; }
  float mu = sm * (1.f / 16.f);
  float var = sq * (1.f / 16.f) - mu * mu;
  float rs = rsqrtf(var + 1e-5f);
  float4* yp = (float4*)(y + (size_t)i * 16);
#pragma unroll
  for (int q = 0; q < 4; ++q) {
    float4 v;
    v.x = fmaxf((r[4 * q + 0] - mu) * rs * g[4 * q + 0] + be[4 * q + 0], 0.f);
    v.y = fmaxf((r[4 * q + 1] - mu) * rs * g[4 * q + 1] + be[4 * q + 1], 0.f);
    v.z = fmaxf((r[4 * q + 2] - mu) * rs * g[4 * q + 2] + be[4 * q + 2], 0.f);
    v.w = fmaxf((r[4 * q + 3] - mu) * rs * g[4 * q + 3] + be[4 * q + 3], 0.f);
    yp[q] = v;
  }
}

// init m (running max keys) over NH elems and sn (s|num interleaved) over 2*NH elems
__global__ void k_init_scatter(int* __restrict__ m_i, float* __restrict__ sn, int nh) {
  int i = blockIdx.x * blockDim.x + threadIdx.x;
  if (i < nh) m_i[i] = INT_MIN;
  if (i < 2 * nh) sn[i] = 0.f;
}

// ---------------- edge passes ----------------
// MODE 0: segment max of z into m_i (float-as-ordered-int keys)
// MODE 1: e = exp(z - m); sn[dst*32+f] += e; sn[dst*32+16+f] += msg*e
template <int MODE>
__global__ void k_edge(const float* __restrict__ edge_attr, const float* __restrict__ ew,
                       const float* __restrict__ ebias, const int* __restrict__ ei,
                       const float* __restrict__ y, const float* __restrict__ tptr, int layer,
                       int* __restrict__ m_i, float* __restrict__ sn, int E) {
  __shared__ float sw[256];
  __shared__ float sb[16];
  int tid = threadIdx.x;
  sw[tid] = ew[tid];
  if (tid < 16) sb[tid] = ebias[tid];
  __syncthreads();
  int e = blockIdx.x * 256 + tid;
  if (e >= E) return;
  float tv = tptr[layer];
  int src = ei[e];
  int dst = ei[E + e];
  float att[16], ea[16], yj[16];
  const float4* ap = (const float4*)(edge_attr + (size_t)e * 16);
  const float4* yp = (const float4*)(y + (size_t)src * 16);
#pragma unroll
  for (int q = 0; q < 4; ++q) {
    float4 va = ap[q];
    float4 vy = yp[q];
    att[4 * q] = va.x; att[4 * q + 1] = va.y; att[4 * q + 2] = va.z; att[4 * q + 3] = va.w;
    yj[4 * q] = vy.x; yj[4 * q + 1] = vy.y; yj[4 * q + 2] = vy.z; yj[4 * q + 3] = vy.w;
  }
#pragma unroll
  for (int f = 0; f < 16; ++f) ea[f] = sb[f];
#pragma unroll
  for (int k = 0; k < 16; ++k) {
    float av = att[k];
#pragma unroll
    for (int f = 0; f < 16; ++f) ea[f] = fmaf(av, sw[k * 16 + f], ea[f]);
  }
  size_t mbase = (size_t)dst * 16;
  size_t sbase = (size_t)dst * 32;
#pragma unroll
  for (int f = 0; f < 16; ++f) {
    float msg = fmaxf(yj[f] + ea[f], 0.f) + EPSM;
    float z = tv * msg;
    if (MODE == 0) {
      atomicMax(&m_i[mbase + f], f2key(z));
    } else {
      float em = __expf(z - key2f(m_i[mbase + f]));
      atomicAdd(&sn[sbase + f], em);            // denominator s
      atomicAdd(&sn[sbase + 16 + f], msg * em); // numerator sum(msg*e)
    }
  }
}

// ---------------- per-layer MLP: h' = Lin2(relu(LN(Lin1(y + num/s)))) (+ residual) ----------------
__global__ void k_mlp(const float* __restrict__ y, const float* __restrict__ sn,
                      float* __restrict__ h, const float* __restrict__ w1,
                      const float* __restrict__ b1, const float* __restrict__ g1,
                      const float* __restrict__ be1, const float* __restrict__ w2,
                      const float* __restrict__ b2, int is0, int ntiles) {
  __shared__ float lds[10][16][33];
  int wid = threadIdx.x >> 5;
  int gw = (int)((blockIdx.x * blockDim.x + threadIdx.x) >> 5);
  int lane = threadIdx.x & 31;
  int lrow = lane & 15;
  int half = lane >> 4;
  int col = lane & 15;
  if (gw >= ntiles) return;                 // wave-uniform
  int row0 = gw << 4;

  // GEMM1: (y + num/(s+1e-16))[16x16] @ w1[16x32] -> two 16x16 tiles, K=16
  v8f c0 = {0.f, 0.f, 0.f, 0.f, 0.f, 0.f, 0.f, 0.f};
  v8f c1 = {0.f, 0.f, 0.f, 0.f, 0.f, 0.f, 0.f, 0.f};
#pragma unroll
  for (int kk = 0; kk < 16; kk += 4) {
    int ka = kk + 2 * half;
    size_t node = (size_t)(row0 + lrow);
    float s0 = sn[node * 32 + ka];
    float n0 = sn[node * 32 + 16 + ka];
    float s1 = sn[node * 32 + ka + 1];
    float n1 = sn[node * 32 + 17 + ka];
    v2f a, b0v, b1v;
    a.x = y[node * 16 + ka] + n0 / (s0 + 1e-16f);
    a.y = y[node * 16 + ka + 1] + n1 / (s1 + 1e-16f);
    b0v.x = w1[(size_t)ka * 32 + col];
    b0v.y = w1[(size_t)(ka + 1) * 32 + col];
    b1v.x = w1[(size_t)ka * 32 + 16 + col];
    b1v.y = w1[(size_t)(ka + 1) * 32 + 16 + col];
    c0 = wmma4(a, b0v, c0);
    c1 = wmma4(a, b1v, c1);
  }
  float bb0 = b1[col], bb1 = b1[16 + col];
  float gg0 = g1[col], gg1 = g1[16 + col];
  float ee0 = be1[col], ee1 = be1[16 + col];
  // LN over 32 cols per row (rows striped: vgpr v -> row v + 8*half; cols across lanes)
#pragma unroll
  for (int v = 0; v < 8; ++v) {
    float u0 = c0[v] + bb0;
    float u1 = c1[v] + bb1;
    float sm = u0 + u1;
    float sq = u0 * u0 + u1 * u1;
#pragma unroll
    for (int m = 1; m <= 8; m <<= 1) {
      sm += __shfl_xor(sm, m, 32);
      sq += __shfl_xor(sq, m, 32);
    }
    float mu = sm * (1.f / 32.f);
    float var = sq * (1.f / 32.f) - mu * mu;
    float rs = rsqrtf(var + 1e-5f);
    u0 = fmaxf((u0 - mu) * rs * gg0 + ee0, 0.f);
    u1 = fmaxf((u1 - mu) * rs * gg1 + ee1, 0.f);
    lds[wid][v + 8 * half][col] = u0;
    lds[wid][v + 8 * half][16 + col] = u1;
  }
  __syncthreads();
  // GEMM2: u[16x32] @ w2[32x16], K=32 (A refetched from LDS in A-layout)
  v8f c2 = {0.f, 0.f, 0.f, 0.f, 0.f, 0.f, 0.f, 0.f};
#pragma unroll
  for (int kk = 0; kk < 32; kk += 4) {
    int ka = kk + 2 * half;
    v2f a, b;
    a.x = lds[wid][lrow][ka];
    a.y = lds[wid][lrow][ka + 1];
    b.x = w2[(size_t)ka * 16 + col];
    b.y = w2[(size_t)(ka + 1) * 16 + col];
    c2 = wmma4(a, b, c2);
  }
  float bv = b2[col];
#pragma unroll
  for (int v = 0; v < 8; ++v) {
    size_t idx = (size_t)(row0 + v + 8 * half) * 16 + col;
    float out = c2[v] + bv;
    h[idx] = is0 ? out : (h[idx] + out);
  }
}

// ---------------- pooling ----------------
__global__ void k_init_pool(int* __restrict__ gmp_i, float* __restrict__ gsum,
                            float* __restrict__ gcnt, int B) {
  int i = blockIdx.x * blockDim.x + threadIdx.x;
  if (i < B * 16) { gmp_i[i] = INT_MIN; gsum[i] = 0.f; }
  if (i < B) gcnt[i] = 0.f;
}

__global__ void k_pool(const float* __restrict__ h, const int* __restrict__ batch,
                       const float* __restrict__ g, const float* __restrict__ be,
                       int* __restrict__ gmp_i, float* __restrict__ gsum,
                       float* __restrict__ gcnt, int n) {
  int i = blockIdx.x * blockDim.x + threadIdx.x;
  if (i >= n) return;
  float r[16];
  const float4* hp = (const float4*)(h + (size_t)i * 16);
#pragma unroll
  for (int q = 0; q < 4; ++q) {
    float4 v = hp[q];
    r[4 * q] = v.x; r[4 * q + 1] = v.y; r[4 * q + 2] = v.z; r[4 * q + 3] = v.w;
  }
  float sm = 0.f, sq = 0.f;
#pragma unroll
  for (int f = 0; f < 16; ++f) { sm += r[f]; sq += r[f] * r[f]; }
  float mu = sm * (1.f / 16.f);
  float var = sq * (1.f / 16.f) - mu * mu;
  float rs = rsqrtf(var + 1e-5f);
  int b = batch[i];
#pragma unroll
  for (int f = 0; f < 16; ++f) {
    float v = fmaxf((r[f] - mu) * rs * g[f] + be[f], 0.f);
    atomicMax(&gmp_i[b * 16 + f], f2key(v));
    atomicAdd(&gsum[b * 16 + f], v);
  }
  atomicAdd(&gcnt[b], 1.f);
}

// ---------------- head: mol -> fp -> pol -> out ----------------
__global__ void k_head(const int* __restrict__ gmp_i, const float* __restrict__ gsum,
                       const float* __restrict__ gcnt, const float* __restrict__ action,
                       const float* __restrict__ pin_w, const float* __restrict__ pin_b,
                       const float* __restrict__ ph_w, const float* __restrict__ ph_b,
                       const float* __restrict__ po_w, const float* __restrict__ po_b,
                       float* __restrict__ out) {
  __shared__ float mol[64 * 32];
  __shared__ float fp[64 * 128];
  int tid = threadIdx.x;
  for (int i = tid; i < 64 * 32; i += 256) {
    int b = i >> 5, f = i & 31;
    mol[i] = (f < 16) ? key2f(gmp_i[b * 16 + f]) : gsum[b * 16 + (f - 16)] / gcnt[b];
  }
  __syncthreads();
  for (int i = tid; i < 64 * 128; i += 256) {
    int b = i >> 7, f = i & 127;
    float acc = pin_b[f];
#pragma unroll 8
    for (int k = 0; k < 32; ++k) acc = fmaf(mol[b * 32 + k], pin_w[k * 128 + f], acc);
    fp[i] = fmaxf(acc, 0.f);
  }
  __syncthreads();
  if (tid < 64) {
    float pol[10];
#pragma unroll
    for (int j = 0; j < 10; ++j) {
      float acc = ph_b[j];
      for (int k = 0; k < 128; ++k) acc = fmaf(fp[tid * 128 + k], ph_w[k * 10 + j], acc);
#pragma unroll
      for (int k = 0; k < 6; ++k) acc = fmaf(action[tid * 6 + k], ph_w[(128 + k) * 10 + j], acc);
      pol[j] = fmaxf(acc, 0.f);
    }
    float o = po_b[0];
#pragma unroll
    for (int j = 0; j < 10; ++j) o = fmaf(pol[j], po_w[j], o);
    out[tid] = o;
  }
}

extern "C" void kernel_launch(void* const* d_in, const int* in_sizes, int n_in,
                              void* d_out, int out_size, void* d_ws, size_t ws_size,
                              hipStream_t stream) {
  const float* x         = (const float*)d_in[0];
  const float* edge_attr = (const float*)d_in[1];
  const int*   edge_index= (const int*)d_in[2];
  const int*   batch     = (const int*)d_in[3];
  const float* action    = (const float*)d_in[4];
  const float* node_w    = (const float*)d_in[5];
  const float* node_b    = (const float*)d_in[6];
  const float* edge_w    = (const float*)d_in[7];
  const float* edge_b    = (const float*)d_in[8];
  const float* tvec      = (const float*)d_in[9];
  const float* mlp_w1    = (const float*)d_in[10];
  const float* mlp_b1    = (const float*)d_in[11];
  const float* mlp_g1    = (const float*)d_in[12];
  const float* mlp_be1   = (const float*)d_in[13];
  const float* mlp_w2    = (const float*)d_in[14];
  const float* mlp_b2    = (const float*)d_in[15];
  const float* ln_g      = (const float*)d_in[16];
  const float* ln_b      = (const float*)d_in[17];
  const float* pin_w     = (const float*)d_in[18];
  const float* pin_b     = (const float*)d_in[19];
  const float* ph_w      = (const float*)d_in[20];
  const float* ph_b      = (const float*)d_in[21];
  const float* po_w      = (const float*)d_in[22];
  const float* po_b      = (const float*)d_in[23];
  float* out = (float*)d_out;

  int N = in_sizes[0] / 64;
  int E = in_sizes[2] / 2;
  int B = in_sizes[4] / 6;
  size_t NH = (size_t)N * 16;

  float* ws   = (float*)d_ws;
  float* h    = ws;                       // [N,16]
  float* y    = ws + NH;                  // [N,16]
  float* sn   = ws + 2 * NH;              // [N,32] interleaved: s | num
  int*   m_i  = (int*)(ws + 4 * NH);      // [N,16] ordered-int max keys
  float* gsum = ws + 5 * NH;
  float* gcnt = gsum + (size_t)B * 16;
  int*   gmp_i = (int*)(gcnt + B);

  int ntiles = N / 16;                       // N = 100000 -> 6250 tiles
  dim3 blkW(320);                            // 10 waves/block
  int grdW = (ntiles + 9) / 10;
  int grdN = (N + 255) / 256;
  int grd2NH = (int)((2 * NH + 255) / 256);
  int grdE = (E + 255) / 256;

  k_node_encode<<<grdW, blkW, 0, stream>>>(x, node_w, node_b, h, ntiles);

  for (int l = 0; l < 4; ++l) {
    const float* yin = h;
    if (l > 0) {
      k_prep_y<<<grdN, 256, 0, stream>>>(h, ln_g + l * 16, ln_b + l * 16, y, N);
      yin = y;
    }
    k_init_scatter<<<grd2NH, 256, 0, stream>>>(m_i, sn, (int)NH);
    k_edge<0><<<grdE, 256, 0, stream>>>(edge_attr, edge_w, edge_b, edge_index, yin,
                                        tvec, l, m_i, sn, E);
    k_edge<1><<<grdE, 256, 0, stream>>>(edge_attr, edge_w, edge_b, edge_index, yin,
                                        tvec, l, m_i, sn, E);
    k_mlp<<<grdW, blkW, 0, stream>>>(yin, sn, h, mlp_w1 + (size_t)l * 16 * 32,
                                     mlp_b1 + l * 32, mlp_g1 + l * 32, mlp_be1 + l * 32,
                                     mlp_w2 + (size_t)l * 32 * 16, mlp_b2 + l * 16,
                                     l == 0 ? 1 : 0, ntiles);
  }

  k_init_pool<<<(B * 16 + 255) / 256, 256, 0, stream>>>(gmp_i, gsum, gcnt, B);
  k_pool<<<grdN, 256, 0, stream>>>(h, batch, ln_g, ln_b, gmp_i, gsum, gcnt, N);
  k_head<<<1, 256, 0, stream>>>(gmp_i, gsum, gcnt, action, pin_w, pin_b, ph_w, ph_b,
                                po_w, po_b, out);
}